// FeatureExtractor_42236708389684
// MI455X (gfx1250) — compile-verified
//
#include <hip/hip_runtime.h>
#include <hip/hip_bf16.h>

typedef __attribute__((ext_vector_type(2))) float v2f;
typedef __attribute__((ext_vector_type(8))) float v8f;

#define HW 12544     // 112*112
#define NEG_BIG (-3.402823466e38f)

// ---------------------------------------------------------------------------
// Kernel 1: per-batch attention map (head-mean, bilinear 16x16 -> 112x112)
//           + iterative top-64 (value desc, index asc tie-break)
// ---------------------------------------------------------------------------
__global__ void attn_topk_kernel(const float* __restrict__ attn, int* __restrict__ topk) {
    __shared__ float av[HW];
    __shared__ float mh[256];
    __shared__ float rv[256];
    __shared__ int   ri[256];

    const int b = blockIdx.x;
    const int tid = threadIdx.x;

    // head-mean of attn[b,:,16,16] (mean commutes with bilinear resize)
    {
        float s = 0.0f;
        #pragma unroll
        for (int h = 0; h < 6; ++h)
            s += attn[(b * 6 + h) * 256 + tid];
        mh[tid] = s * (1.0f / 6.0f);
    }
    __syncthreads();

    // bilinear upsample with half-pixel centers (align_corners=False)
    for (int p = tid; p < HW; p += 256) {
        int y = p / 112, x = p % 112;
        float sy = (y + 0.5f) * (16.0f / 112.0f) - 0.5f;
        float sx = (x + 0.5f) * (16.0f / 112.0f) - 0.5f;
        int y0 = (int)floorf(sy); float fy = sy - (float)y0;
        int x0 = (int)floorf(sx); float fx = sx - (float)x0;
        int y0c = min(15, max(0, y0)), y1c = min(15, max(0, y0 + 1));
        int x0c = min(15, max(0, x0)), x1c = min(15, max(0, x0 + 1));
        float v00 = mh[y0c * 16 + x0c], v01 = mh[y0c * 16 + x1c];
        float v10 = mh[y1c * 16 + x0c], v11 = mh[y1c * 16 + x1c];
        float v0 = v00 + fx * (v01 - v00);
        float v1 = v10 + fx * (v11 - v10);
        av[p] = v0 + fy * (v1 - v0);
    }
    __syncthreads();

    // 64 x argmax-extract
    for (int k = 0; k < 64; ++k) {
        float bv = NEG_BIG; int bi = 0x7fffffff;
        for (int p = tid; p < HW; p += 256) {
            float v = av[p];
            if (v > bv || (v == bv && p < bi)) { bv = v; bi = p; }
        }
        rv[tid] = bv; ri[tid] = bi;
        __syncthreads();
        for (int s = 128; s > 0; s >>= 1) {
            if (tid < s) {
                float ov = rv[tid + s]; int oi = ri[tid + s];
                if (ov > rv[tid] || (ov == rv[tid] && oi < ri[tid])) { rv[tid] = ov; ri[tid] = oi; }
            }
            __syncthreads();
        }
        if (tid == 0) {
            topk[b * 64 + k] = ri[0];
            av[ri[0]] = NEG_BIG;
        }
        __syncthreads();
    }
}

// ---------------------------------------------------------------------------
// Kernel 2: per-batch  proj = ReLU(BN(W @ combined + bias)) at the 64 selected
// pixels only, via V_WMMA_F32_16X16X4_F32 (M=128, N=64, K=512), then column
// L2 normalization.  One workgroup (8 wave32) per batch; wave w owns M-tile w.
//
// B-operand is staged in LDS as K-PAIRS: pairB[pi][n] = (B[2pi][n], B[2pi+1][n])
// so each WMMA B fragment is one aligned ds_load_b64 (no register shuffles).
// Pair-row stride 80 float2s => upper half-wave (pi+1) hits banks 32..63,
// lower half-wave banks 0..31: conflict-free b64 loads.
// ---------------------------------------------------------------------------
#define PSTR 80          // pair-row stride in float2 units (2*80 = 32 mod 64 banks)
#define SMS  65          // float stride for the reused P[128][64] view

__global__ void proj_topk_kernel(const float* __restrict__ lf,
                                 const float* __restrict__ pt,
                                 const float* __restrict__ convW,
                                 const float* __restrict__ convB,
                                 const float* __restrict__ gamma,
                                 const float* __restrict__ beta,
                                 const float* __restrict__ mean,
                                 const float* __restrict__ var,
                                 const int*   __restrict__ topk,
                                 float* __restrict__ out) {
    // 64 pair-rows * 80 float2 = 40 KB; reused afterwards as P[128][65] floats (33 KB)
    __shared__ __align__(16) float smem[64 * PSTR * 2];
    __shared__ int   pix[64];
    __shared__ int   tok[64][4];
    __shared__ float wgt[64][4];
    __shared__ float chA[128];          // BN scale
    __shared__ float chB[128];          // fused bias*scale + shift
    __shared__ float invn[64];

    v2f* pairB = (v2f*)smem;

    const int b   = blockIdx.x;
    const int tid = threadIdx.x;
    const int wave  = tid >> 5;
    const int lane  = tid & 31;
    const int lhalf = lane & 15;
    const int hi    = lane >> 4;        // 0 for lanes 0-15 (K-pair 0), 1 for 16-31 (K-pair 1)

    // per-pixel bilinear gather descriptors (token indices + weights)
    if (tid < 64) {
        int p = topk[b * 64 + tid];
        pix[tid] = p;
        int y = p / 112, x = p % 112;
        float sy = (y + 0.5f) * (16.0f / 112.0f) - 0.5f;
        float sx = (x + 0.5f) * (16.0f / 112.0f) - 0.5f;
        int y0 = (int)floorf(sy); float fy = sy - (float)y0;
        int x0 = (int)floorf(sx); float fx = sx - (float)x0;
        int y0c = min(15, max(0, y0)), y1c = min(15, max(0, y0 + 1));
        int x0c = min(15, max(0, x0)), x1c = min(15, max(0, x0 + 1));
        tok[tid][0] = y0c * 16 + x0c;  tok[tid][1] = y0c * 16 + x1c;
        tok[tid][2] = y1c * 16 + x0c;  tok[tid][3] = y1c * 16 + x1c;
        wgt[tid][0] = (1.0f - fy) * (1.0f - fx);  wgt[tid][1] = (1.0f - fy) * fx;
        wgt[tid][2] = fy * (1.0f - fx);           wgt[tid][3] = fy * fx;
    }
    if (tid < 128) {
        float sc = gamma[tid] * rsqrtf(var[tid] + 1e-5f);
        float sh = beta[tid] - mean[tid] * sc;
        chA[tid] = sc;
        chB[tid] = convB[tid] * sc + sh;
    }

    v8f acc0 = {}, acc1 = {}, acc2 = {}, acc3 = {};

    const float* ptB = pt + (size_t)b * 256 * 384;
    const float* lfB = lf + (size_t)b * 128 * HW;

    for (int kc = 0; kc < 4; ++kc) {
        __syncthreads();
        // build combined K-chunk as 64 K-pair rows x 64 pixels (one b64 store each)
        for (int e = tid; e < 64 * 64; e += 256) {
            int pi = e >> 6, n = e & 63;
            int kRow = kc * 128 + pi * 2;            // even; pair never straddles 384
            v2f v;
            if (kc < 3) {                            // patch-token rows (kRow+1 < 384)
                v2f t0 = *(const v2f*)(ptB + tok[n][0] * 384 + kRow);
                v2f t1 = *(const v2f*)(ptB + tok[n][1] * 384 + kRow);
                v2f t2 = *(const v2f*)(ptB + tok[n][2] * 384 + kRow);
                v2f t3 = *(const v2f*)(ptB + tok[n][3] * 384 + kRow);
                v.x = wgt[n][0]*t0.x + wgt[n][1]*t1.x + wgt[n][2]*t2.x + wgt[n][3]*t3.x;
                v.y = wgt[n][0]*t0.y + wgt[n][1]*t1.y + wgt[n][2]*t2.y + wgt[n][3]*t3.y;
            } else {                                 // local-feature rows (kRow >= 384)
                int c = kRow - 384;
                v.x = lfB[(size_t)(c    ) * HW + pix[n]];
                v.y = lfB[(size_t)(c + 1) * HW + pix[n]];
            }
            pairB[pi * PSTR + n] = v;
        }
        __syncthreads();

        // WMMA over this chunk: wave owns M-tile = wave, N-tiles 0..3
        const float* wr = convW + (size_t)(wave * 16 + lhalf) * 512 + kc * 128 + 2 * hi;
        #pragma unroll 4
        for (int ks = 0; ks < 32; ++ks) {
            v2f a = *(const v2f*)(wr + ks * 4);      // A[m][k+2hi], A[m][k+2hi+1]
            int prow = (ks * 2 + hi) * PSTR;         // K-pair row for this half-wave
            v2f b0 = pairB[prow +  0 + lhalf];
            v2f b1 = pairB[prow + 16 + lhalf];
            v2f b2 = pairB[prow + 32 + lhalf];
            v2f b3 = pairB[prow + 48 + lhalf];
            acc0 = __builtin_amdgcn_wmma_f32_16x16x4_f32(false, a, false, b0, (short)0, acc0, false, false);
            acc1 = __builtin_amdgcn_wmma_f32_16x16x4_f32(false, a, false, b1, (short)0, acc1, false, false);
            acc2 = __builtin_amdgcn_wmma_f32_16x16x4_f32(false, a, false, b2, (short)0, acc2, false, false);
            acc3 = __builtin_amdgcn_wmma_f32_16x16x4_f32(false, a, false, b3, (short)0, acc3, false, false);
        }
    }
    __syncthreads();

    // fused bias + BN + ReLU, stage P[128][64] in (reused) LDS
    {
        int rowbase = wave * 16 + ((lane < 16) ? 0 : 8);
        #pragma unroll
        for (int r = 0; r < 8; ++r) {
            int row = rowbase + r;
            float sc = chA[row], sh = chB[row];
            smem[row * SMS +  0 + lhalf] = fmaxf(acc0[r] * sc + sh, 0.0f);
            smem[row * SMS + 16 + lhalf] = fmaxf(acc1[r] * sc + sh, 0.0f);
            smem[row * SMS + 32 + lhalf] = fmaxf(acc2[r] * sc + sh, 0.0f);
            smem[row * SMS + 48 + lhalf] = fmaxf(acc3[r] * sc + sh, 0.0f);
        }
    }
    __syncthreads();

    // column L2 norms over the 128 channels
    if (tid < 64) {
        float s = 0.0f;
        #pragma unroll 4
        for (int c = 0; c < 128; ++c) {
            float v = smem[c * SMS + tid];
            s += v * v;
        }
        invn[tid] = 1.0f / fmaxf(sqrtf(s), 1e-12f);
    }
    __syncthreads();

    // out[b][c][k] = P[c][k] * invn[k]
    float* outB = out + (size_t)b * 128 * 64;
    for (int e = tid; e < 128 * 64; e += 256) {
        int c = e >> 6, k = e & 63;
        outB[e] = smem[c * SMS + k] * invn[k];
    }
}

// ---------------------------------------------------------------------------
extern "C" void kernel_launch(void* const* d_in, const int* in_sizes, int n_in,
                              void* d_out, int out_size, void* d_ws, size_t ws_size,
                              hipStream_t stream) {
    (void)in_sizes; (void)n_in; (void)out_size; (void)ws_size;
    const float* lf    = (const float*)d_in[0];  // (16,128,112,112)
    const float* pt    = (const float*)d_in[1];  // (16,256,384)
    const float* attn  = (const float*)d_in[2];  // (16,6,16,16)
    const float* convW = (const float*)d_in[3];  // (128,512)
    const float* convB = (const float*)d_in[4];  // (128,)
    const float* gamma = (const float*)d_in[5];
    const float* beta  = (const float*)d_in[6];
    const float* mean  = (const float*)d_in[7];
    const float* var   = (const float*)d_in[8];
    float* out = (float*)d_out;                  // (16,128,64)
    int* topk = (int*)d_ws;                      // 16*64 ints of scratch

    attn_topk_kernel<<<16, 256, 0, stream>>>(attn, topk);
    proj_topk_kernel<<<16, 256, 0, stream>>>(lf, pt, convW, convB, gamma, beta,
                                             mean, var, topk, out);
}